// SelfAttentionBlock_283467841719
// MI455X (gfx1250) — compile-verified
//
#include <hip/hip_runtime.h>

typedef float v2f __attribute__((ext_vector_type(2)));
typedef float v8f __attribute__((ext_vector_type(8)));

#define BB 16
#define LL 2048
#define DD 192
#define HH 4
#define KK 12
#define NROWS (BB * LL)   // 32768
#define QKVN 48           // H*K

__device__ __forceinline__ v8f wmma4(v2f a, v2f b, v8f c) {
  // D = A(16x4,f32) * B(4x16,f32) + C(16x16,f32)
  return __builtin_amdgcn_wmma_f32_16x16x4_f32(false, a, false, b, (short)0, c,
                                               false, false);
}

// ---------------- LayerNorm: one wave (32 lanes) per row of 192 ----------------
__global__ void ln_kernel(const float* __restrict__ x, const float* __restrict__ g,
                          const float* __restrict__ bta, float* __restrict__ y) {
  const int lane = threadIdx.x & 31;
  const int row = (blockIdx.x * blockDim.x + threadIdx.x) >> 5;
  if (row >= NROWS) return;
  const float* xr = x + (size_t)row * DD;
  float v[6];
  float s = 0.f;
#pragma unroll
  for (int i = 0; i < 6; ++i) { v[i] = xr[lane + 32 * i]; s += v[i]; }
#pragma unroll
  for (int off = 16; off; off >>= 1) s += __shfl_xor(s, off, 32);
  const float mean = s * (1.0f / DD);
  float q = 0.f;
#pragma unroll
  for (int i = 0; i < 6; ++i) { float d = v[i] - mean; q += d * d; }
#pragma unroll
  for (int off = 16; off; off >>= 1) q += __shfl_xor(q, off, 32);
  const float r = rsqrtf(q * (1.0f / DD) + 1e-3f);
  float* yr = y + (size_t)row * DD;
#pragma unroll
  for (int i = 0; i < 6; ++i) {
    int c = lane + 32 * i;
    yr[c] = (v[i] - mean) * r * g[c] + bta[c];
  }
}

// ------------- QKV projection: wave per (mtile, q/k/v, ntile) GEMM tile -------------
__global__ void qkv_kernel(const float* __restrict__ xn,
                           const float* __restrict__ wq, const float* __restrict__ bq,
                           const float* __restrict__ wk, const float* __restrict__ bk,
                           const float* __restrict__ wv, const float* __restrict__ bv,
                           float* __restrict__ qo, float* __restrict__ ko,
                           float* __restrict__ vo) {
  const int gw = (blockIdx.x * blockDim.x + threadIdx.x) >> 5;
  const int lane = threadIdx.x & 31, half = lane >> 4, li = lane & 15;
  const int mtile = gw / 9, r = gw - mtile * 9;
  const int which = r / 3, ntile = r - which * 3;
  const float *W, *Bv;
  float* O;
  if (which == 0) { W = wq; Bv = bq; O = qo; }
  else if (which == 1) { W = wk; Bv = bk; O = ko; }
  else { W = wv; Bv = bv; O = vo; }
  const float* A = xn + (size_t)mtile * 16 * DD;
  const int nc = ntile * 16 + li;
  v8f c = {};
  for (int k0 = 0; k0 < DD; k0 += 4) {
    v2f a, b;
    a.x = A[li * DD + k0 + 2 * half];
    a.y = A[li * DD + k0 + 2 * half + 1];
    b.x = W[(k0 + 2 * half) * QKVN + nc];
    b.y = W[(k0 + 2 * half + 1) * QKVN + nc];
    c = wmma4(a, b, c);
  }
  const float bias = Bv[nc];
#pragma unroll
  for (int m = 0; m < 8; ++m)
    O[((size_t)mtile * 16 + m + 8 * half) * QKVN + nc] = c[m] + bias;
}

// ------------- Flash-style causal attention: wave per (b, h, 16-query tile) -------------
__global__ void attn_kernel(const float* __restrict__ q, const float* __restrict__ k,
                            const float* __restrict__ v, float* __restrict__ ctx) {
  __shared__ float lds[8 * 16 * 17];  // per-wave padded 16x16 P tile
  const int gw = (blockIdx.x * blockDim.x + threadIdx.x) >> 5;
  const int lane = threadIdx.x & 31, half = lane >> 4, li = lane & 15;
  float* P = lds + (threadIdx.x >> 5) * (16 * 17);
  const int b = gw >> 9;           // / (H*128)
  const int h = (gw >> 7) & 3;
  const int qt = gw & 127;
  const size_t rowbase = (size_t)b * LL;
  const float* qp = q + (rowbase + qt * 16) * QKVN + h * KK;
  const float scale = 0.28867513459481287f;  // 1/sqrt(K)
  v2f qa[3];
#pragma unroll
  for (int s3 = 0; s3 < 3; ++s3) {
    int k0 = s3 * 4;
    qa[s3].x = qp[li * QKVN + k0 + 2 * half] * scale;
    qa[s3].y = qp[li * QKVN + k0 + 2 * half + 1] * scale;
  }
  v8f acc = {};
  float mrow[8], srow[8];
#pragma unroll
  for (int m = 0; m < 8; ++m) { mrow[m] = -3.0e38f; srow[m] = 0.f; }
  for (int j = 0; j <= qt; ++j) {
    const float* kp = k + (rowbase + j * 16) * QKVN + h * KK;
    v8f s = {};
#pragma unroll
    for (int s3 = 0; s3 < 3; ++s3) {
      int k0 = s3 * 4;
      v2f bf;
      bf.x = kp[li * QKVN + k0 + 2 * half];
      bf.y = kp[li * QKVN + k0 + 2 * half + 1];
      s = wmma4(qa[s3], bf, s);
    }
    if (j == qt) {  // additive causal mask inside diagonal tile
#pragma unroll
      for (int m = 0; m < 8; ++m)
        if (li > m + 8 * half) s[m] -= 1.0e9f;
    }
#pragma unroll
    for (int m = 0; m < 8; ++m) {  // online softmax; row lives in a 16-lane half
      float rm = s[m];
      rm = fmaxf(rm, __shfl_xor(rm, 1, 32));
      rm = fmaxf(rm, __shfl_xor(rm, 2, 32));
      rm = fmaxf(rm, __shfl_xor(rm, 4, 32));
      rm = fmaxf(rm, __shfl_xor(rm, 8, 32));
      float mn = fmaxf(mrow[m], rm);
      float corr = __expf(mrow[m] - mn);
      mrow[m] = mn;
      float p = __expf(s[m] - mn);
      float rs = p;
      rs += __shfl_xor(rs, 1, 32);
      rs += __shfl_xor(rs, 2, 32);
      rs += __shfl_xor(rs, 4, 32);
      rs += __shfl_xor(rs, 8, 32);
      srow[m] = srow[m] * corr + rs;
      acc[m] *= corr;
      P[(m + 8 * half) * 17 + li] = p;  // C-layout -> LDS for A-layout reload
    }
    const float* vp = v + (rowbase + j * 16) * QKVN + h * KK;
#pragma unroll
    for (int s4 = 0; s4 < 4; ++s4) {  // acc += P(16x16) * V(16x12)
      int k0 = s4 * 4;
      v2f a, bf;
      a.x = P[li * 17 + k0 + 2 * half];
      a.y = P[li * 17 + k0 + 2 * half + 1];
      bf.x = (li < KK) ? vp[(k0 + 2 * half) * QKVN + li] : 0.f;
      bf.y = (li < KK) ? vp[(k0 + 2 * half + 1) * QKVN + li] : 0.f;
      acc = wmma4(a, bf, acc);
    }
  }
  if (li < KK) {
    float* op = ctx + (rowbase + qt * 16) * QKVN + h * KK;
#pragma unroll
    for (int m = 0; m < 8; ++m)
      op[(m + 8 * half) * QKVN + li] = acc[m] / srow[m];
  }
}

// ------------- Output projection + residual: wave per (mtile, ntile) -------------
__global__ void outproj_kernel(const float* __restrict__ ctx, const float* __restrict__ wo,
                               const float* __restrict__ bo, const float* __restrict__ x,
                               float* __restrict__ x2) {
  const int gw = (blockIdx.x * blockDim.x + threadIdx.x) >> 5;
  const int lane = threadIdx.x & 31, half = lane >> 4, li = lane & 15;
  const int mtile = gw / 12, ntile = gw - mtile * 12;
  const float* A = ctx + (size_t)mtile * 16 * QKVN;
  const int nc = ntile * 16 + li;
  v8f c = {};
#pragma unroll
  for (int k0 = 0; k0 < QKVN; k0 += 4) {
    v2f a, b;
    a.x = A[li * QKVN + k0 + 2 * half];
    a.y = A[li * QKVN + k0 + 2 * half + 1];
    b.x = wo[(k0 + 2 * half) * DD + nc];
    b.y = wo[(k0 + 2 * half + 1) * DD + nc];
    c = wmma4(a, b, c);
  }
  const float bias = bo[nc];
#pragma unroll
  for (int m = 0; m < 8; ++m) {
    size_t row = (size_t)mtile * 16 + m + 8 * half;
    x2[row * DD + nc] = x[row * DD + nc] + c[m] + bias;
  }
}

// ------------- 3-tap valid conv as implicit GEMM (K=3*192); border rows zeroed -------------
__global__ void conv3_kernel(const float* __restrict__ in, const float* __restrict__ w,
                             const float* __restrict__ bias, const float* __restrict__ resid,
                             float* __restrict__ out, int do_relu) {
  const int gw = (blockIdx.x * blockDim.x + threadIdx.x) >> 5;
  const int lane = threadIdx.x & 31, half = lane >> 4, li = lane & 15;
  const int mtile = gw / 12, ntile = gw - mtile * 12;
  const int nc = ntile * 16 + li;
  const int g0 = mtile * 16;
  v8f c = {};
  for (int t = 0; t < 3; ++t) {
    int r = g0 + li + t - 1;                       // border outputs are zeroed below,
    r = r < 0 ? 0 : (r > NROWS - 1 ? NROWS - 1 : r);  // so clamped reads are harmless
    const float* Ar = in + (size_t)r * DD;
    const float* Wt = w + (size_t)t * DD * DD;
    for (int k0 = 0; k0 < DD; k0 += 4) {
      v2f a, b;
      a.x = Ar[k0 + 2 * half];
      a.y = Ar[k0 + 2 * half + 1];
      b.x = Wt[(k0 + 2 * half) * DD + nc];
      b.y = Wt[(k0 + 2 * half + 1) * DD + nc];
      c = wmma4(a, b, c);
    }
  }
  const float bv = bias[nc];
#pragma unroll
  for (int m = 0; m < 8; ++m) {
    int g = g0 + m + 8 * half;
    int l = g & (LL - 1);
    float val = c[m] + bv;
    if (do_relu) val = fmaxf(val, 0.f);
    if (l == 0 || l == LL - 1) val = 0.f;   // VALID conv + ZeroPadding1D(1,1)
    if (resid) val += resid[(size_t)g * DD + nc];
    out[(size_t)g * DD + nc] = val;
  }
}

extern "C" void kernel_launch(void* const* d_in, const int* in_sizes, int n_in,
                              void* d_out, int out_size, void* d_ws, size_t ws_size,
                              hipStream_t stream) {
  (void)in_sizes; (void)n_in; (void)out_size; (void)ws_size;
  const float* x    = (const float*)d_in[0];
  const float* ln1g = (const float*)d_in[1];
  const float* ln1b = (const float*)d_in[2];
  const float* wq   = (const float*)d_in[3];
  const float* bq   = (const float*)d_in[4];
  const float* wk   = (const float*)d_in[5];
  const float* bk   = (const float*)d_in[6];
  const float* wv   = (const float*)d_in[7];
  const float* bv   = (const float*)d_in[8];
  const float* wo   = (const float*)d_in[9];
  const float* bo   = (const float*)d_in[10];
  const float* ln2g = (const float*)d_in[11];
  const float* ln2b = (const float*)d_in[12];
  const float* c1w  = (const float*)d_in[13];
  const float* c1b  = (const float*)d_in[14];
  const float* c2w  = (const float*)d_in[15];
  const float* c2b  = (const float*)d_in[16];
  float* out = (float*)d_out;
  float* ws = (float*)d_ws;

  // Workspace (floats): [0,6.29M) xn/xn2 ; [6.29M,12.58M) q|k|v|ctx (reused as hp) ;
  // [12.58M,18.87M) x2.  Total 72 MB.
  float* xn  = ws;
  float* qb  = ws + 6291456;
  float* kb  = ws + 7864320;
  float* vb  = ws + 9437184;
  float* ctx = ws + 11010048;
  float* hp  = qb;  // q/k/v/ctx dead after out-proj; exactly 6291456 floats
  float* x2  = ws + 12582912;

  dim3 blk(256);  // 8 wave32 per workgroup
  ln_kernel<<<4096, blk, 0, stream>>>(x, ln1g, ln1b, xn);
  qkv_kernel<<<2304, blk, 0, stream>>>(xn, wq, bq, wk, bk, wv, bv, qb, kb, vb);
  attn_kernel<<<1024, blk, 0, stream>>>(qb, kb, vb, ctx);
  outproj_kernel<<<3072, blk, 0, stream>>>(ctx, wo, bo, x, x2);
  ln_kernel<<<4096, blk, 0, stream>>>(x2, ln2g, ln2b, xn);
  conv3_kernel<<<3072, blk, 0, stream>>>(xn, c1w, c1b, nullptr, hp, 1);
  conv3_kernel<<<3072, blk, 0, stream>>>(hp, c2w, c2b, x2, out, 0);
}